// SmoothLDDTLoss_23124103922400
// MI455X (gfx1250) — compile-verified
//
#include <hip/hip_runtime.h>
#include <hip/hip_bf16.h>
#include <math.h>

typedef float v2f __attribute__((ext_vector_type(2)));
typedef float v8f __attribute__((ext_vector_type(8)));

// One wave (32 lanes) computes one 16x16 tile of the pairwise matrices.
// Only tiles with tn >= tm are computed; off-diagonal tiles are weighted 2x
// (eps and the mask are symmetric in (i,j)).
//
// Distance trick: A row m = [-2x, -2y, -2z, |x_m|^2], B col n = [x, y, z, 1]
//   => WMMA D[m][n] = |x_m|^2 - 2 x_m.x_n ;  d^2 = D[m][n] + |x_n|^2,
// where |x_n|^2 is already resident in the lane that owns column n.
__global__ void __launch_bounds__(256)
lddt_tile_kernel(const float* __restrict__ pred,
                 const float* __restrict__ trueC,
                 const int*   __restrict__ is_dna,
                 const int*   __restrict__ is_rna,
                 float*       __restrict__ acc,   // [2*b]: {sum, count} per batch
                 int n, int T, long long triPerBatch, long long totalTiles)
{
    const int wave = threadIdx.x >> 5;
    const int lane = threadIdx.x & 31;
    const long long tile_id = (long long)blockIdx.x * 8 + wave;
    if (tile_id >= totalTiles) return;   // whole-wave uniform exit; EXEC stays full

    const int batch = (int)(tile_id / triPerBatch);
    const int t     = (int)(tile_id % triPerBatch);

    // Triangular decomposition: tm such that off(tm) <= t < off(tm+1),
    // off(tm) = tm*T - tm*(tm-1)/2. Float seed + exact integer fixup.
    int tm = (int)((2.0f * T + 1.0f -
                    sqrtf((2.0f * T + 1.0f) * (2.0f * T + 1.0f) - 8.0f * (float)t)) * 0.5f);
    if (tm < 0) tm = 0;
    if (tm > T - 1) tm = T - 1;
    while (tm > 0 && (tm * T - tm * (tm - 1) / 2) > t) --tm;
    while (((tm + 1) * T - (tm + 1) * tm / 2) <= t) ++tm;
    const int tn = tm + (t - (tm * T - tm * (tm - 1) / 2));

    const int i0 = tm << 4;
    const int j0 = tn << 4;
    const int lr = lane & 15;
    const bool hi = lane >= 16;

    const size_t baseB = (size_t)batch * n;
    const float* pri = pred  + (baseB + i0 + lr) * 3;
    const float* prj = pred  + (baseB + j0 + lr) * 3;
    const float* tri = trueC + (baseB + i0 + lr) * 3;
    const float* trj = trueC + (baseB + j0 + lr) * 3;

    const float pix = pri[0], piy = pri[1], piz = pri[2];
    const float pjx = prj[0], pjy = prj[1], pjz = prj[2];
    const float tix = tri[0], tiy = tri[1], tiz = tri[2];
    const float tjx = trj[0], tjy = trj[1], tjz = trj[2];

    // squared norms: lane L holds sq for row i0+(L&15) / col j0+(L&15)
    const float sqPi = pix*pix + piy*piy + piz*piz;
    const float sqPj = pjx*pjx + pjy*pjy + pjz*pjz;
    const float sqTi = tix*tix + tiy*tiy + tiz*tiz;
    const float sqTj = tjx*tjx + tjy*tjy + tjz*tjz;

    // A (16x4): lanes 0-15 -> (K0,K1) = (-2x,-2y); lanes 16-31 -> (K2,K3) = (-2z, |x|^2)
    // B (4x16): lanes 0-15 -> (K0,K1) = ( x,  y);  lanes 16-31 -> (K2,K3) = (  z, 1)
    v2f aP, bP, aT, bT;
    aP.x = hi ? (-2.0f * piz) : (-2.0f * pix);  aP.y = hi ? sqPi : (-2.0f * piy);
    bP.x = hi ? pjz : pjx;                      bP.y = hi ? 1.0f : pjy;
    aT.x = hi ? (-2.0f * tiz) : (-2.0f * tix);  aT.y = hi ? sqTi : (-2.0f * tiy);
    bT.x = hi ? tjz : tjx;                      bT.y = hi ? 1.0f : tjy;

    v8f czero = {};
    // gP[m][nn] = |x_{i0+m}|^2 - 2 x_{i0+m} . x_{j0+nn}
    v8f gP = __builtin_amdgcn_wmma_f32_16x16x4_f32(false, aP, false, bP,
                                                   (short)0, czero, false, false);
    v8f gT = __builtin_amdgcn_wmma_f32_16x16x4_f32(false, aT, false, bT,
                                                   (short)0, czero, false, false);

    // nucleic flags: row flags broadcast via wave32 ballot (bit m = row i0+m)
    const int nucI = (is_dna[baseB + i0 + lr] | is_rna[baseB + i0 + lr]) ? 1 : 0;
    const int nucJ = (is_dna[baseB + j0 + lr] | is_rna[baseB + j0 + lr]) ? 1 : 0;
    const unsigned rowNucMask = __builtin_amdgcn_ballot_w32(nucI != 0);

    const int mBase = hi ? 8 : 0;
    const int offdiagI = (tm != tn) ? 1 : 0;     // bitwise, no short-circuit

    // sigm(c - diff) = 1 / (1 + e^{diff} * e^{-c})
    const float K1 = 0.60653066f;   // e^-0.5
    const float K2 = 0.36787944f;   // e^-1
    const float K3 = 0.13533528f;   // e^-2
    const float K4 = 0.018315639f;  // e^-4

    float sum = 0.0f;
    int   cnt = 0;
#pragma unroll
    for (int v = 0; v < 8; ++v) {
        const int m = v + mBase;                 // row within tile for this VGPR

        const float d2p = fmaxf(gP[v] + sqPj, 1e-12f);
        const float d2t = fmaxf(gT[v] + sqTj, 1e-12f);
        const float dp = __builtin_amdgcn_sqrtf(d2p);
        const float dt = __builtin_amdgcn_sqrtf(d2t);
        const float diff = fabsf(dt - dp);

        const float E = __expf(diff);            // one v_exp per pair
        const float eps4 = __builtin_amdgcn_rcpf(1.0f + E * K1)
                         + __builtin_amdgcn_rcpf(1.0f + E * K2)
                         + __builtin_amdgcn_rcpf(1.0f + E * K3)
                         + __builtin_amdgcn_rcpf(1.0f + E * K4);

        // all-bitwise predicate: no EXEC branches
        const int nucPair = (int)((rowNucMask >> m) & 1u) & nucJ;
        const float cutoff = nucPair ? 30.0f : 15.0f;
        const int inc = ((int)(dt < cutoff)) & (offdiagI | (int)(m != lr));

        const float msk = inc ? 1.0f : 0.0f;
        cnt += inc;
        sum = fmaf(msk, eps4, sum);
    }

    // hoisted uniform scaling: eps 1/4 factor and the 2x symmetry weight
    const float w = (tm == tn) ? 1.0f : 2.0f;
    float fsum = sum * (0.25f * w);
    float fcnt = (float)cnt * w;

    // wave32 reduction
#pragma unroll
    for (int off = 16; off > 0; off >>= 1) {
        fsum += __shfl_down(fsum, off, 32);
        fcnt += __shfl_down(fcnt, off, 32);
    }
    if (lane == 0) {
        atomicAdd(&acc[2 * batch + 0], fsum);
        atomicAdd(&acc[2 * batch + 1], fcnt);
    }
}

__global__ void lddt_finalize(const float* __restrict__ acc,
                              float* __restrict__ out, int b)
{
    if (threadIdx.x == 0 && blockIdx.x == 0) {
        float s = 0.0f;
        for (int i = 0; i < b; ++i) {
            const float c = fmaxf(acc[2 * i + 1], 1.0f);
            s += acc[2 * i] / c;
        }
        out[0] = 1.0f - s / (float)b;
    }
}

extern "C" void kernel_launch(void* const* d_in, const int* in_sizes, int n_in,
                              void* d_out, int out_size, void* d_ws, size_t ws_size,
                              hipStream_t stream) {
    const float* pred  = (const float*)d_in[0];
    const float* truec = (const float*)d_in[1];
    const int*   isdna = (const int*)d_in[2];
    const int*   isrna = (const int*)d_in[3];

    const int b = 2;                        // reference: b=2
    const int n = in_sizes[0] / (3 * b);    // 4096
    const int T = n / 16;                   // 256 tiles per row
    const long long triPerBatch = (long long)T * (T + 1) / 2;
    const long long totalTiles  = triPerBatch * b;
    const int blocks = (int)((totalTiles + 7) / 8);   // 8 waves per 256-thread block

    float* acc = (float*)d_ws;
    hipMemsetAsync(acc, 0, 2 * b * sizeof(float), stream);

    lddt_tile_kernel<<<blocks, 256, 0, stream>>>(pred, truec, isdna, isrna,
                                                 acc, n, T, triPerBatch, totalTiles);
    lddt_finalize<<<1, 32, 0, stream>>>(acc, (float*)d_out, b);
}